// Adapter_79190607003730
// MI455X (gfx1250) — compile-verified
//
#include <hip/hip_runtime.h>
#include <stdint.h>

// Fused adapter: LayerNorm -> down(768->64)+ReLU -> up(64->768)+bias.
// fp32 end-to-end using V_WMMA_F32_16X16X4_F32 (wave32, gfx1250).
// Memory-bound: one HBM read of x (async-copied straight into LDS), one HBM
// write of out; all intermediates stay in LDS.

typedef float v2f __attribute__((ext_vector_type(2)));
typedef float v8f __attribute__((ext_vector_type(8)));

#define D_MODEL   768
#define BOT       64
#define TILE_ROWS 32          // rows per block
#define THREADS   256         // 8 waves (wave32)
#define H_STRIDE  772         // 768 + 4 pad: column reads hit distinct LDS banks
#define DT_STRIDE 65          // 64 + 1 pad
#define LN_EPS    1e-5f

__launch_bounds__(THREADS, 1)
__global__ void adapter_fused(const float* __restrict__ x,
                              const float* __restrict__ gamma,
                              const float* __restrict__ beta,
                              const float* __restrict__ w_down,  // [768][64]
                              const float* __restrict__ b_down,  // [64]
                              const float* __restrict__ w_up,    // [64][768]
                              const float* __restrict__ b_up,    // [768]
                              float* __restrict__ out)
{
    __shared__ __align__(16) float h[TILE_ROWS][H_STRIDE];    // normalized activations
    __shared__ __align__(16) float dt[TILE_ROWS][DT_STRIDE];  // relu(down) tile
    __shared__ float redS[THREADS];
    __shared__ float redQ[THREADS];
    __shared__ float sMean[TILE_ROWS];
    __shared__ float sRstd[TILE_ROWS];

    const int tid  = threadIdx.x;
    const int lane = tid & 31;        // wave32
    const int wave = tid >> 5;        // 0..7
    const int half = lane >> 4;       // 0/1 (K-half select for A/B frags)
    const int l16  = lane & 15;       // M (A rows) / N (B,C cols)
    const long row0 = (long)blockIdx.x * TILE_ROWS;
    const float* xblk = x + row0 * D_MODEL;
    float* oblk = out + row0 * D_MODEL;

    // ---------- Phase 1: async-copy x tile straight into LDS ----------
    // 32*768 floats = 6144 float4 transfers / 256 threads = 24 per thread.
    // global_load_async_to_lds_b128: no VGPR landing, tracked by ASYNCcnt.
    {
        const uint32_t lds_base = (uint32_t)(uintptr_t)&h[0][0];
        #pragma unroll
        for (int i = 0; i < (TILE_ROWS * D_MODEL / 4) / THREADS; ++i) {
            const int f  = i * THREADS + tid;          // float4 index in tile
            const int r  = f / (D_MODEL / 4);
            const int c4 = (f % (D_MODEL / 4)) * 4;
            const float*   gsrc = xblk + (long)f * 4;
            const uint32_t ldst = lds_base +
                (uint32_t)((r * H_STRIDE + c4) * sizeof(float)); // 16B aligned
            asm volatile("global_load_async_to_lds_b128 %0, %1, off"
                         :: "v"(ldst), "v"(gsrc)
                         : "memory");
        }
        asm volatile("s_wait_asynccnt 0x0" ::: "memory");
    }
    __syncthreads();

    // ---------- Phase 2: LayerNorm (8 threads per row, 96 elems each) ----------
    {
        const int r = tid >> 3;
        const int p = tid & 7;
        float s = 0.f, q = 0.f;
        const int c0 = p * (D_MODEL / 8);
        #pragma unroll 8
        for (int c = c0; c < c0 + (D_MODEL / 8); ++c) {
            const float v = h[r][c];
            s += v; q += v * v;
        }
        redS[tid] = s; redQ[tid] = q;
    }
    __syncthreads();
    if (tid < TILE_ROWS) {
        float s = 0.f, q = 0.f;
        #pragma unroll
        for (int i = 0; i < 8; ++i) { s += redS[tid * 8 + i]; q += redQ[tid * 8 + i]; }
        const float mean = s * (1.0f / D_MODEL);
        const float var  = q * (1.0f / D_MODEL) - mean * mean;   // biased, like jnp.var
        sMean[tid] = mean;
        sRstd[tid] = rsqrtf(var + LN_EPS);
    }
    __syncthreads();
    // normalize + affine, in place
    #pragma unroll 4
    for (int i = 0; i < (TILE_ROWS * D_MODEL) / THREADS; ++i) {
        const int f = i * THREADS + tid;
        const int r = f / D_MODEL;
        const int c = f % D_MODEL;
        const float v = h[r][c];
        h[r][c] = (v - sMean[r]) * sRstd[r] * gamma[c] + beta[c];
    }
    __syncthreads();

    // ---------- Phase 3: GEMM1  h[32x768] @ w_down[768x64] (+bias, ReLU) ----------
    // 2 M-tiles x 4 N-tiles -> one 16x16 wave-tile per wave.
    {
        const int mt = wave >> 2;          // 0..1
        const int nt = wave & 3;           // 0..3
        const float* wd = w_down + nt * 16 + l16;   // column base for this lane
        const int arow = mt * 16 + l16;
        v8f acc0 = {}; v8f acc1 = {};      // two chains -> independent WMMA streams
        #pragma unroll 4
        for (int ks = 0; ks < D_MODEL / 8; ++ks) {
            // chain 0: K-block 8*ks .. +3
            {
                const int kb = ks * 8 + 2 * half;
                v2f a, b;
                a[0] = h[arow][kb + 0];
                a[1] = h[arow][kb + 1];
                b[0] = wd[(kb + 0) * BOT];
                b[1] = wd[(kb + 1) * BOT];
                acc0 = __builtin_amdgcn_wmma_f32_16x16x4_f32(
                           false, a, false, b, (short)0, acc0, false, false);
            }
            // chain 1: K-block 8*ks+4 .. +7
            {
                const int kb = ks * 8 + 4 + 2 * half;
                v2f a, b;
                a[0] = h[arow][kb + 0];
                a[1] = h[arow][kb + 1];
                b[0] = wd[(kb + 0) * BOT];
                b[1] = wd[(kb + 1) * BOT];
                acc1 = __builtin_amdgcn_wmma_f32_16x16x4_f32(
                           false, a, false, b, (short)0, acc1, false, false);
            }
        }
        const float bd = b_down[nt * 16 + l16];
        #pragma unroll
        for (int r = 0; r < 8; ++r) {
            const float v = (acc0[r] + acc1[r]) + bd;
            dt[mt * 16 + r + 8 * half][nt * 16 + l16] = v > 0.f ? v : 0.f;
        }
    }
    __syncthreads();

    // ---------- Phase 4: GEMM2  dt[32x64] @ w_up[64x768] (+bias) ----------
    // 48 N-tiles / 8 waves = 6 per wave; B frags register-resident per N-tile,
    // both M-tiles interleaved as independent accumulator chains.
    for (int i = 0; i < (D_MODEL / 16) / 8; ++i) {
        const int nt = wave + 8 * i;                   // 0..47
        const float* wu = w_up + nt * 16 + l16;        // column base for this lane
        v2f b[BOT / 4];
        #pragma unroll
        for (int ks = 0; ks < BOT / 4; ++ks) {
            const int kb = ks * 4 + 2 * half;
            b[ks][0] = wu[(kb + 0) * D_MODEL];
            b[ks][1] = wu[(kb + 1) * D_MODEL];
        }
        const float bu = b_up[nt * 16 + l16];

        v8f acc0 = {}; v8f acc1 = {};
        #pragma unroll
        for (int ks = 0; ks < BOT / 4; ++ks) {
            const int kb = ks * 4 + 2 * half;
            v2f a0, a1;
            a0[0] = dt[l16][kb + 0];
            a0[1] = dt[l16][kb + 1];
            a1[0] = dt[16 + l16][kb + 0];
            a1[1] = dt[16 + l16][kb + 1];
            acc0 = __builtin_amdgcn_wmma_f32_16x16x4_f32(
                       false, a0, false, b[ks], (short)0, acc0, false, false);
            acc1 = __builtin_amdgcn_wmma_f32_16x16x4_f32(
                       false, a1, false, b[ks], (short)0, acc1, false, false);
        }
        #pragma unroll
        for (int r = 0; r < 8; ++r) {
            const int m = r + 8 * half;
            oblk[(long)(m)      * D_MODEL + nt * 16 + l16] = acc0[r] + bu;
            oblk[(long)(m + 16) * D_MODEL + nt * 16 + l16] = acc1[r] + bu;
        }
    }
}

extern "C" void kernel_launch(void* const* d_in, const int* in_sizes, int n_in,
                              void* d_out, int out_size, void* d_ws, size_t ws_size,
                              hipStream_t stream) {
    (void)n_in; (void)out_size; (void)d_ws; (void)ws_size;
    const float* x      = (const float*)d_in[0];
    const float* gamma  = (const float*)d_in[1];
    const float* beta   = (const float*)d_in[2];
    const float* w_down = (const float*)d_in[3];
    const float* b_down = (const float*)d_in[4];
    const float* w_up   = (const float*)d_in[5];
    const float* b_up   = (const float*)d_in[6];
    float* out = (float*)d_out;

    const int n_rows = in_sizes[0] / D_MODEL;     // 8*4096 = 32768
    const int grid   = n_rows / TILE_ROWS;        // 1024 blocks
    adapter_fused<<<grid, THREADS, 0, stream>>>(x, gamma, beta, w_down, b_down,
                                                w_up, b_up, out);
}